// GCN_layers_28226525069659
// MI455X (gfx1250) — compile-verified
//
#include <hip/hip_runtime.h>
#include <hip/hip_bf16.h>
#include <math.h>

// Problem constants (match reference)
#define BB 64
#define NN 512
#define DD 768
#define LN_EPS 1e-5f

typedef float v2f __attribute__((ext_vector_type(2)));
typedef float v8f __attribute__((ext_vector_type(8)));

// ---------------------------------------------------------------------------
// Stage 1a: partial[c][b][d] = sum over n in chunk c of mask[b,n]*sent[b,n,d]
// grid (64, 3, 4), block 256.  Deterministic (each block writes its own slot).
// ---------------------------------------------------------------------------
__global__ void masked_partial_sum(const float* __restrict__ sent,
                                   const int* __restrict__ mask,
                                   float* __restrict__ partial) {
    const int b = blockIdx.x;                       // batch
    const int d = blockIdx.y * 256 + threadIdx.x;   // 0..767
    const int c = blockIdx.z;                       // n-chunk 0..3
    const int n0 = c * 128;

    __shared__ float sm[128];
    if (threadIdx.x < 128)
        sm[threadIdx.x] = (float)mask[b * NN + n0 + threadIdx.x];
    __syncthreads();

    const float* base = sent + ((size_t)b * NN + n0) * DD + d;
    float acc = 0.0f;
#pragma unroll 8
    for (int n = 0; n < 128; ++n)
        acc += sm[n] * base[(size_t)n * DD];

    partial[((size_t)c * BB + b) * DD + d] = acc;
}

// ---------------------------------------------------------------------------
// Stage 1b: s[i] = sum_c partial[c][i]   (i over 64*768)
// ---------------------------------------------------------------------------
__global__ void reduce_partials(const float* __restrict__ partial,
                                float* __restrict__ s) {
    const int i = blockIdx.x * 256 + threadIdx.x;
    const int STRIDE = BB * DD;
    s[i] = partial[i] + partial[STRIDE + i] + partial[2 * STRIDE + i] +
           partial[3 * STRIDE + i];
}

// ---------------------------------------------------------------------------
// cnt[b] = number of active nodes in batch b
// ---------------------------------------------------------------------------
__global__ void cnt_kernel(const int* __restrict__ mask,
                           float* __restrict__ cnt) {
    const int b = threadIdx.x;
    if (b < BB) {
        int c = 0;
        for (int n = 0; n < NN; ++n) c += mask[b * NN + n];
        cnt[b] = (float)c;
    }
}

// ---------------------------------------------------------------------------
// voff[d] = tanh(b2[d])   (output rows for inactive nodes)
// ---------------------------------------------------------------------------
__global__ void voff_kernel(const float* __restrict__ b2,
                            float* __restrict__ voff) {
    const int d = blockIdx.x * 256 + threadIdx.x;
    if (d < DD) voff[d] = tanhf(b2[d]);
}

// ---------------------------------------------------------------------------
// out[m,n] = tanh( (A @ W)[m,n] + bias[n] ) * (useScale ? rowScale[m] : 1)
// A: [64,768], W: [768,768].  One wave (32 threads) per 16x16 output tile,
// fp32 WMMA 16x16x4, K-loop of 192 iterations.
//   A frag (16x4 f32): lanes 0-15 hold rows M=lane, K={k,k+1} in v[0],v[1];
//                      lanes 16-31 hold rows M=lane-16, K={k+2,k+3}.
//   B frag (4x16 f32): lanes 0-15 hold cols N=lane for K={k,k+1};
//                      lanes 16-31 cols N=lane-16 for K={k+2,k+3}.
//   C/D (16x16 f32):   lanes 0-15: vgpr i -> (M=i,     N=lane)
//                      lanes 16-31: vgpr i -> (M=i+8,  N=lane-16)
// ---------------------------------------------------------------------------
__global__ void gemm_tanh(const float* __restrict__ A,
                          const float* __restrict__ W,
                          const float* __restrict__ bias,
                          const float* __restrict__ rowScale,
                          float* __restrict__ out, int useScale) {
    const int lane = threadIdx.x;        // 0..31
    const int half = lane >> 4;          // 0 or 1
    const int idx  = lane & 15;
    const int m0 = blockIdx.x << 4;
    const int n0 = blockIdx.y << 4;

    const float* arow = A + (m0 + idx) * DD;   // this lane's A row
    const float* wcol = W + n0 + idx;          // this lane's B column base

    v8f acc = {};
    for (int k = 0; k < DD; k += 4) {
        const int ka = k + (half << 1);
        v2f a;  a.x  = arow[ka];          a.y  = arow[ka + 1];
        v2f bf; bf.x = wcol[ka * DD];     bf.y = wcol[(ka + 1) * DD];
        acc = __builtin_amdgcn_wmma_f32_16x16x4_f32(
            false, a, false, bf, (short)0, acc, false, false);
    }

    const int n = n0 + idx;
    const float bv = bias[n];
#pragma unroll
    for (int i = 0; i < 8; ++i) {
        const int m = m0 + (half << 3) + i;
        float v = tanhf(acc[i] + bv);
        if (useScale) v *= rowScale[m];
        out[m * DD + n] = v;
    }
}

// ---------------------------------------------------------------------------
// Final pass: x = sent[row,:] + (mask[row] ? von[b,:] : voff[:]); LayerNorm.
// One 192-thread block (6 waves) per row; float4 per thread (768 = 192*4).
// ---------------------------------------------------------------------------
__global__ void residual_ln(const float* __restrict__ sent,
                            const int* __restrict__ mask,
                            const float* __restrict__ von,
                            const float* __restrict__ voff,
                            const float* __restrict__ gamma,
                            const float* __restrict__ beta,
                            float* __restrict__ out) {
    const int row = blockIdx.x;          // 0..B*N-1
    const int b   = row >> 9;            // N == 512
    const int tid = threadIdx.x;         // 0..191

    const float4* srow = (const float4*)(sent + (size_t)row * DD);
    const float*  vsrc = mask[row] ? (von + (size_t)b * DD) : voff;
    const float4* vrow = (const float4*)vsrc;

    float4 x = srow[tid];
    float4 v = vrow[tid];
    x.x += v.x; x.y += v.y; x.z += v.z; x.w += v.w;

    float sum = x.x + x.y + x.z + x.w;
    float sq  = x.x * x.x + x.y * x.y + x.z * x.z + x.w * x.w;

    // wave32 reduction
    for (int o = 16; o > 0; o >>= 1) {
        sum += __shfl_xor(sum, o, 32);
        sq  += __shfl_xor(sq,  o, 32);
    }

    __shared__ float ssum[6], ssq[6];
    const int w = tid >> 5, l = tid & 31;
    if (l == 0) { ssum[w] = sum; ssq[w] = sq; }
    __syncthreads();
    if (tid == 0) {
        float a = 0.0f, c = 0.0f;
        for (int i = 0; i < 6; ++i) { a += ssum[i]; c += ssq[i]; }
        ssum[0] = a; ssq[0] = c;
    }
    __syncthreads();

    const float tot = ssum[0], totsq = ssq[0];
    const float mu   = tot * (1.0f / DD);
    const float var  = totsq * (1.0f / DD) - mu * mu;
    const float rstd = rsqrtf(var + LN_EPS);

    const float4 g  = ((const float4*)gamma)[tid];
    const float4 be = ((const float4*)beta)[tid];
    float4 y;
    y.x = (x.x - mu) * rstd * g.x + be.x;
    y.y = (x.y - mu) * rstd * g.y + be.y;
    y.z = (x.z - mu) * rstd * g.z + be.z;
    y.w = (x.w - mu) * rstd * g.w + be.w;
    ((float4*)(out + (size_t)row * DD))[tid] = y;
}

// ---------------------------------------------------------------------------
extern "C" void kernel_launch(void* const* d_in, const int* in_sizes, int n_in,
                              void* d_out, int out_size, void* d_ws,
                              size_t ws_size, hipStream_t stream) {
    (void)in_sizes; (void)n_in; (void)out_size; (void)ws_size;

    const float* sent  = (const float*)d_in[0];
    const int*   mask  = (const int*)  d_in[1];
    const float* W1    = (const float*)d_in[2];
    const float* b1    = (const float*)d_in[3];
    const float* W2    = (const float*)d_in[4];
    const float* b2    = (const float*)d_in[5];
    const float* gamma = (const float*)d_in[6];
    const float* beta  = (const float*)d_in[7];
    float* out = (float*)d_out;

    // Workspace layout (floats); total ~1.38 MB
    float* ws      = (float*)d_ws;
    float* partial = ws;                     // 4*64*768
    float* s       = partial + 4 * BB * DD;  // 64*768
    float* cnt     = s + BB * DD;            // 64
    float* s2      = cnt + 64;               // 64*768
    float* von     = s2 + BB * DD;           // 64*768
    float* voff    = von + BB * DD;          // 768

    // 1) masked reduction over N (reads sent once: 96 MB)
    masked_partial_sum<<<dim3(BB, 3, 4), 256, 0, stream>>>(sent, mask, partial);
    cnt_kernel<<<1, 64, 0, stream>>>(mask, cnt);
    voff_kernel<<<3, 256, 0, stream>>>(b2, voff);
    reduce_partials<<<(BB * DD) / 256, 256, 0, stream>>>(partial, s);

    // 2) t1 = s @ W1 ; s2 = cnt * tanh(t1 + b1)      (WMMA fp32)
    gemm_tanh<<<dim3(BB / 16, DD / 16), 32, 0, stream>>>(s, W1, b1, cnt, s2, 1);
    // 3) t2 = s2 @ W2 ; von = tanh(t2 + b2)           (WMMA fp32)
    gemm_tanh<<<dim3(BB / 16, DD / 16), 32, 0, stream>>>(s2, W2, b2, cnt, von, 0);

    // 4) residual + LayerNorm (reads sent again, writes out: 192 MB)
    residual_ln<<<BB * NN, 192, 0, stream>>>(sent, mask, von, voff, gamma, beta,
                                             out);
}